// Basic_RT_45681272160598
// MI455X (gfx1250) — compile-verified
//
#include <hip/hip_runtime.h>
#include <hip/hip_bf16.h>

// ---------------------------------------------------------------------------
// Graph-transformer block for MI455X (gfx1250, wave32, WMMA).
// fp32 GEMMs on V_WMMA_F32_16X16X4_F32. B staged *transposed* in LDS so both
// A and B fragments are single aligned ds_load_b64 ops (no VGPR repacking).
// Triple-projection GEMMs share the A operand; NL2/EL2 fuse residual+LN.
// ---------------------------------------------------------------------------

typedef __attribute__((ext_vector_type(2))) float v2f;
typedef __attribute__((ext_vector_type(4))) float v4f;
typedef __attribute__((ext_vector_type(8))) float v8f;

#define NS 256
#define KC 32
#define PAD 36   // 144B rows: 16B-aligned, conflict-free b64 fragment reads

// ---------------------------------------------------------------------------
// Single GEMM: Out[M,256] = act(A[M,256] @ W[256,256] + bias). 4 waves,
// 64x64 tile, grid (M/64, 4).
// ---------------------------------------------------------------------------
__global__ __launch_bounds__(128) void gemm256_wmma_kernel(
    const float* __restrict__ A, const float* __restrict__ W,
    const float* __restrict__ bias, float* __restrict__ Out, int relu)
{
  __shared__ float As[64 * PAD];   // [m][k]
  __shared__ float Bt[64 * PAD];   // [n][k]  (transposed at staging)

  const int tid = threadIdx.x, wave = tid >> 5, lane = tid & 31;
  const int lane16 = lane & 15, half = lane >> 4;
  const int m0 = blockIdx.x * 64, n0 = blockIdx.y * 64;

  v8f acc[4];
#pragma unroll
  for (int nt = 0; nt < 4; ++nt)
#pragma unroll
    for (int r = 0; r < 8; ++r) acc[nt][r] = 0.0f;

  for (int kc = 0; kc < NS; kc += KC) {
    __syncthreads();
#pragma unroll
    for (int it = 0; it < 4; ++it) {                  // A 64x32 -> [m][k]
      int idx = tid + it * 128, row = idx >> 3, c4 = (idx & 7) << 2;
      *(v4f*)(As + row * PAD + c4) =
          *(const v4f*)(A + (size_t)(m0 + row) * NS + kc + c4);
    }
#pragma unroll
    for (int it = 0; it < 4; ++it) {                  // B 32x64 -> [n][k]
      int idx = tid + it * 128, row = idx >> 4, c4 = (idx & 15) << 2;
      v4f bv = *(const v4f*)(W + (size_t)(kc + row) * NS + n0 + c4);
      Bt[(c4 + 0) * PAD + row] = bv.x;
      Bt[(c4 + 1) * PAD + row] = bv.y;
      Bt[(c4 + 2) * PAD + row] = bv.z;
      Bt[(c4 + 3) * PAD + row] = bv.w;
    }
    if (kc + KC < NS) {                               // prefetch next chunk
      __builtin_prefetch(A + (size_t)(m0 + (tid & 63)) * NS + kc + KC, 0, 3);
      __builtin_prefetch(W + (size_t)(kc + KC + (tid & 31)) * NS + n0, 0, 3);
    }
    __syncthreads();
#pragma unroll
    for (int k = 0; k < KC; k += 4) {
      v2f a = *(const v2f*)(As + (wave * 16 + lane16) * PAD + k + half * 2);
#pragma unroll
      for (int nt = 0; nt < 4; ++nt) {
        v2f b = *(const v2f*)(Bt + (nt * 16 + lane16) * PAD + k + half * 2);
        acc[nt] = __builtin_amdgcn_wmma_f32_16x16x4_f32(
            false, a, false, b, (short)0, acc[nt], false, false);
      }
    }
  }
#pragma unroll
  for (int nt = 0; nt < 4; ++nt) {
    int n = n0 + nt * 16 + lane16;
    float bv = bias[n];
#pragma unroll
    for (int r = 0; r < 8; ++r) {
      int m = m0 + wave * 16 + r + half * 8;
      float v = acc[nt][r] + bv;
      if (relu) v = fmaxf(v, 0.0f);
      Out[(size_t)m * NS + n] = v;
    }
  }
}

// ---------------------------------------------------------------------------
// Triple GEMM (shared A): O{0,1,2}[M,256] = A @ W3[w] + b3[w].
// A tile staged once per K-chunk, 12 accumulator tiles, 96 WMMA / chunk.
// ---------------------------------------------------------------------------
__global__ __launch_bounds__(128) void gemm256x3_wmma_kernel(
    const float* __restrict__ A, const float* __restrict__ W3,
    const float* __restrict__ b3, float* __restrict__ O0,
    float* __restrict__ O1, float* __restrict__ O2)
{
  __shared__ float As[64 * PAD];
  __shared__ float Bt[3][64 * PAD];                   // [w][n][k]

  const int tid = threadIdx.x, wave = tid >> 5, lane = tid & 31;
  const int lane16 = lane & 15, half = lane >> 4;
  const int m0 = blockIdx.x * 64, n0 = blockIdx.y * 64;

  v8f acc[3][4];
#pragma unroll
  for (int w = 0; w < 3; ++w)
#pragma unroll
    for (int nt = 0; nt < 4; ++nt)
#pragma unroll
      for (int r = 0; r < 8; ++r) acc[w][nt][r] = 0.0f;

  for (int kc = 0; kc < NS; kc += KC) {
    __syncthreads();
#pragma unroll
    for (int it = 0; it < 4; ++it) {                  // A 64x32 once
      int idx = tid + it * 128, row = idx >> 3, c4 = (idx & 7) << 2;
      *(v4f*)(As + row * PAD + c4) =
          *(const v4f*)(A + (size_t)(m0 + row) * NS + kc + c4);
    }
#pragma unroll
    for (int it = 0; it < 12; ++it) {                 // 3x B 32x64 -> [n][k]
      int idx = tid + it * 128;                       // 0..1535
      int w   = idx >> 9;
      int rem = idx & 511;
      int row = rem >> 4, c4 = (rem & 15) << 2;
      v4f bv = *(const v4f*)(W3 + (size_t)w * NS * NS +
                             (size_t)(kc + row) * NS + n0 + c4);
      float* B = &Bt[w][0];
      B[(c4 + 0) * PAD + row] = bv.x;
      B[(c4 + 1) * PAD + row] = bv.y;
      B[(c4 + 2) * PAD + row] = bv.z;
      B[(c4 + 3) * PAD + row] = bv.w;
    }
    if (kc + KC < NS)                                 // prefetch next A chunk
      __builtin_prefetch(A + (size_t)(m0 + (tid & 63)) * NS + kc + KC, 0, 3);
    __syncthreads();
#pragma unroll
    for (int k = 0; k < KC; k += 4) {
      v2f a = *(const v2f*)(As + (wave * 16 + lane16) * PAD + k + half * 2);
#pragma unroll
      for (int w = 0; w < 3; ++w) {
#pragma unroll
        for (int nt = 0; nt < 4; ++nt) {
          v2f b = *(const v2f*)(&Bt[w][0] + (nt * 16 + lane16) * PAD + k + half * 2);
          acc[w][nt] = __builtin_amdgcn_wmma_f32_16x16x4_f32(
              false, a, false, b, (short)0, acc[w][nt], false, false);
        }
      }
    }
  }
  float* const outs[3] = {O0, O1, O2};
#pragma unroll
  for (int w = 0; w < 3; ++w) {
    float* Ow = outs[w];
#pragma unroll
    for (int nt = 0; nt < 4; ++nt) {
      int n = n0 + nt * 16 + lane16;
      float bv = b3[w * NS + n];
#pragma unroll
      for (int r = 0; r < 8; ++r) {
        int m = m0 + wave * 16 + r + half * 8;
        Ow[(size_t)m * NS + n] = acc[w][nt][r] + bv;
      }
    }
  }
}

// ---------------------------------------------------------------------------
// GEMM + residual LayerNorm fusion (full 256-wide row per block):
// Out = LN(X + (A @ W + bias)) * g + beta.  16 acc tiles per wave; row stats
// via intra-half-wave shuffles (C/D rows live in 16-lane halves). grid (M/64).
// ---------------------------------------------------------------------------
__global__ __launch_bounds__(128) void gemm256_ln_wmma_kernel(
    const float* __restrict__ A, const float* __restrict__ W,
    const float* __restrict__ bias, const float* __restrict__ X,
    const float* __restrict__ g, const float* __restrict__ beta,
    float* __restrict__ Out)
{
  __shared__ float As[64 * PAD];
  __shared__ float Bt[256 * PAD];                     // [n][k], n = 0..255

  const int tid = threadIdx.x, wave = tid >> 5, lane = tid & 31;
  const int lane16 = lane & 15, half = lane >> 4;
  const int m0 = blockIdx.x * 64;

  v8f acc[16];
#pragma unroll
  for (int nt = 0; nt < 16; ++nt)
#pragma unroll
    for (int r = 0; r < 8; ++r) acc[nt][r] = 0.0f;

  for (int kc = 0; kc < NS; kc += KC) {
    __syncthreads();
#pragma unroll
    for (int it = 0; it < 4; ++it) {                  // A 64x32
      int idx = tid + it * 128, row = idx >> 3, c4 = (idx & 7) << 2;
      *(v4f*)(As + row * PAD + c4) =
          *(const v4f*)(A + (size_t)(m0 + row) * NS + kc + c4);
    }
#pragma unroll
    for (int it = 0; it < 16; ++it) {                 // B 32x256 -> [n][k]
      int idx = tid + it * 128, row = idx >> 6, c4 = (idx & 63) << 2;
      v4f bv = *(const v4f*)(W + (size_t)(kc + row) * NS + c4);
      Bt[(c4 + 0) * PAD + row] = bv.x;
      Bt[(c4 + 1) * PAD + row] = bv.y;
      Bt[(c4 + 2) * PAD + row] = bv.z;
      Bt[(c4 + 3) * PAD + row] = bv.w;
    }
    if (kc + KC < NS)
      __builtin_prefetch(A + (size_t)(m0 + (tid & 63)) * NS + kc + KC, 0, 3);
    __syncthreads();
#pragma unroll
    for (int k = 0; k < KC; k += 4) {
      v2f a = *(const v2f*)(As + (wave * 16 + lane16) * PAD + k + half * 2);
#pragma unroll
      for (int nt = 0; nt < 16; ++nt) {
        v2f b = *(const v2f*)(Bt + (nt * 16 + lane16) * PAD + k + half * 2);
        acc[nt] = __builtin_amdgcn_wmma_f32_16x16x4_f32(
            false, a, false, b, (short)0, acc[nt], false, false);
      }
    }
  }

  // acc += bias + residual X  (each lane owns column n = nt*16 + lane16)
#pragma unroll
  for (int nt = 0; nt < 16; ++nt) {
    int n = nt * 16 + lane16;
    float bv = bias[n];
#pragma unroll
    for (int r = 0; r < 8; ++r) {
      int m = m0 + wave * 16 + r + half * 8;
      acc[nt][r] += bv + X[(size_t)m * NS + n];
    }
  }

  // per-row mean/var across the 16 lanes of each half-wave, then normalize
#pragma unroll
  for (int r = 0; r < 8; ++r) {
    float s = 0.0f;
#pragma unroll
    for (int nt = 0; nt < 16; ++nt) s += acc[nt][r];
#pragma unroll
    for (int off = 8; off > 0; off >>= 1) s += __shfl_xor(s, off, 32);
    float mean = s * (1.0f / 256.0f);
    float vs = 0.0f;
#pragma unroll
    for (int nt = 0; nt < 16; ++nt) { float d = acc[nt][r] - mean; vs += d * d; }
#pragma unroll
    for (int off = 8; off > 0; off >>= 1) vs += __shfl_xor(vs, off, 32);
    float inv = rsqrtf(vs * (1.0f / 256.0f) + 1e-5f);

    int m = m0 + wave * 16 + r + half * 8;
    float* op = Out + (size_t)m * NS;
#pragma unroll
    for (int nt = 0; nt < 16; ++nt) {
      int n = nt * 16 + lane16;
      op[n] = (acc[nt][r] - mean) * inv * g[n] + beta[n];
    }
  }
}

// ---------------------------------------------------------------------------
// Node attention: one block per (b,h,i). scores over j, softmax, V-weighted sum.
// ---------------------------------------------------------------------------
__global__ __launch_bounds__(128) void node_attn_kernel(
    const float* __restrict__ eQ, const float* __restrict__ eK,
    const float* __restrict__ eV, const float* __restrict__ nQ,
    const float* __restrict__ nK, const float* __restrict__ nV,
    float* __restrict__ attw)
{
  const float scale = 0.17677669529663687f; // 1/sqrt(32)
  int bi = blockIdx.x;
  int i = bi % 96, t1 = bi / 96, h = t1 % 8, b = t1 / 8;
  const int tid = threadIdx.x, lane = tid & 31;

  __shared__ float sc[96];
  __shared__ float red[2];

  size_t rowi    = (size_t)(b * 96 + i);
  size_t base_ij = rowi * 96 * NS + (size_t)h * 32;
  size_t base_i  = rowi * NS + (size_t)h * 32;

  if (tid < 96) {
    const float* q  = eQ + base_ij + (size_t)tid * NS;
    const float* k  = eK + base_ij + (size_t)tid * NS;
    const float* qn = nQ + base_i;
    const float* kn = nK + (size_t)(b * 96 + tid) * NS + (size_t)h * 32;
    float s = 0.0f;
#pragma unroll
    for (int c = 0; c < 32; ++c) s += (q[c] + qn[c]) * (k[c] + kn[c]);
    sc[tid] = s * scale;
  }
  __syncthreads();

  if (tid < 32) {
    float m = -3.0e38f;
    for (int j = lane; j < 96; j += 32) m = fmaxf(m, sc[j]);
#pragma unroll
    for (int off = 16; off > 0; off >>= 1) m = fmaxf(m, __shfl_xor(m, off, 32));
    float s = 0.0f;
    for (int j = lane; j < 96; j += 32) s += __expf(sc[j] - m);
#pragma unroll
    for (int off = 16; off > 0; off >>= 1) s += __shfl_xor(s, off, 32);
    if (lane == 0) { red[0] = m; red[1] = 1.0f / s; }
  }
  __syncthreads();

  if (tid < 96) sc[tid] = __expf(sc[tid] - red[0]) * red[1];
  __syncthreads();

  if (tid < 32) {
    float accv = 0.0f;
    for (int j = 0; j < 96; ++j) {
      float v = eV[base_ij + (size_t)j * NS + lane] +
                nV[(size_t)(b * 96 + j) * NS + (size_t)h * 32 + lane];
      accv += sc[j] * v;
    }
    attw[base_i + lane] = accv;
  }
}

// ---------------------------------------------------------------------------
// Edge update: one block per edge row; wave == head. out = silu(se)*Ve in place.
// ---------------------------------------------------------------------------
__global__ __launch_bounds__(256) void edge_update_kernel(
    const float* __restrict__ Qe, const float* __restrict__ Ke,
    float* __restrict__ Ve_io,
    const float* __restrict__ n1Q, const float* __restrict__ n1K,
    const float* __restrict__ n1V, const float* __restrict__ n2Q,
    const float* __restrict__ n2K, const float* __restrict__ n2V)
{
  const float scale = 0.17677669529663687f;
  size_t row = blockIdx.x;
  int j = (int)(row % 96);
  size_t t = row / 96;
  int i = (int)(t % 96), b = (int)(t / 96);

  int h = threadIdx.x >> 5, c = threadIdx.x & 31;
  int ns = h * 32 + c;

  size_t ri = (size_t)(b * 96 + i) * NS + ns;
  size_t rj = (size_t)(b * 96 + j) * NS + ns;
  size_t e  = row * NS + ns;

  float q = Qe[e] + n1Q[ri] + n2Q[rj];
  float k = Ke[e] + n1K[ri] + n2K[rj];
  float p = q * k;
#pragma unroll
  for (int off = 16; off > 0; off >>= 1) p += __shfl_xor(p, off, 32);
  float se = p * scale;
  float a  = se / (1.0f + __expf(-se));
  float v  = Ve_io[e] + n1V[ri] + n2V[rj];
  Ve_io[e] = a * v;
}

// ---------------------------------------------------------------------------
extern "C" void kernel_launch(void* const* d_in, const int* in_sizes, int n_in,
                              void* d_out, int out_size, void* d_ws, size_t ws_size,
                              hipStream_t stream) {
  (void)in_sizes; (void)n_in; (void)out_size; (void)ws_size;

  const float* node   = (const float*)d_in[0];
  const float* edge   = (const float*)d_in[1];
  const float* Wn_w   = (const float*)d_in[2];
  const float* Wn_b   = (const float*)d_in[3];
  const float* We_w   = (const float*)d_in[4];
  const float* We_b   = (const float*)d_in[5];
  const float* Wn1_w  = (const float*)d_in[6];
  const float* Wn1_b  = (const float*)d_in[7];
  const float* Wn2_w  = (const float*)d_in[8];
  const float* Wn2_b  = (const float*)d_in[9];
  const float* Wee_w  = (const float*)d_in[10];
  const float* Wee_b  = (const float*)d_in[11];
  const float* NL1_w  = (const float*)d_in[12];
  const float* NL1_b  = (const float*)d_in[13];
  const float* NL2_w  = (const float*)d_in[14];
  const float* NL2_b  = (const float*)d_in[15];
  const float* EL1_w  = (const float*)d_in[16];
  const float* EL1_b  = (const float*)d_in[17];
  const float* EL2_w  = (const float*)d_in[18];
  const float* EL2_b  = (const float*)d_in[19];
  const float* NLN1_g = (const float*)d_in[20];
  const float* NLN1_b = (const float*)d_in[21];
  const float* ELN1_g = (const float*)d_in[22];
  const float* ELN1_b = (const float*)d_in[23];

  const int ME = 73728;                 // B*N*N edge rows
  const int MN = 768;                   // B*N node rows
  const size_t ESZ = (size_t)ME * NS;
  const size_t NSZ = (size_t)MN * NS;

  float* out_node = (float*)d_out;      // [768,256]
  float* out_edge = out_node + NSZ;     // [73728,256]

  float* ws0 = (float*)d_ws;
  float* ws1 = ws0 + ESZ;
  float* ws2 = ws1 + ESZ;
  float* nb  = ws2 + ESZ;
  float* nQ  = nb + 0 * NSZ;  float* nK  = nb + 1 * NSZ;  float* nV  = nb + 2 * NSZ;
  float* atw = nb + 3 * NSZ;  float* h1  = nb + 4 * NSZ;
  float* n1Q = nb + 5 * NSZ;  float* n1K = nb + 6 * NSZ;  float* n1V = nb + 7 * NSZ;
  float* n2Q = nb + 8 * NSZ;  float* n2K = nb + 9 * NSZ;  float* n2V = nb + 10 * NSZ;

  dim3 gE(ME / 64, 4), gN(MN / 64, 4);
  dim3 gEL(ME / 64), gNL(MN / 64);
  const int TB = 128;

  // ---------------- Phase A: node attention ----------------
  gemm256x3_wmma_kernel<<<gE, TB, 0, stream>>>(edge, We_w, We_b, ws0, ws1, ws2);
  gemm256x3_wmma_kernel<<<gN, TB, 0, stream>>>(node, Wn_w, Wn_b, nQ, nK, nV);

  node_attn_kernel<<<8 * 8 * 96, 128, 0, stream>>>(ws0, ws1, ws2, nQ, nK, nV, atw);

  gemm256_wmma_kernel<<<gN, TB, 0, stream>>>(atw, NL1_w, NL1_b, h1, 1);
  gemm256_ln_wmma_kernel<<<gNL, TB, 0, stream>>>(h1, NL2_w, NL2_b,
                                                 node, NLN1_g, NLN1_b, out_node);

  // ---------------- Phase B: edge update ----------------
  gemm256x3_wmma_kernel<<<gN, TB, 0, stream>>>(out_node, Wn1_w, Wn1_b, n1Q, n1K, n1V);
  gemm256x3_wmma_kernel<<<gN, TB, 0, stream>>>(out_node, Wn2_w, Wn2_b, n2Q, n2K, n2V);
  gemm256x3_wmma_kernel<<<gE, TB, 0, stream>>>(edge, Wee_w, Wee_b, ws0, ws1, ws2);

  edge_update_kernel<<<ME, 256, 0, stream>>>(ws0, ws1, ws2, n1Q, n1K, n1V, n2Q, n2K, n2V);

  gemm256_wmma_kernel<<<gE, TB, 0, stream>>>(ws2, EL1_w, EL1_b, ws0, 1);
  gemm256_ln_wmma_kernel<<<gEL, TB, 0, stream>>>(ws0, EL2_w, EL2_b,
                                                 edge, ELN1_g, ELN1_b, out_edge);
}